// MANN_LSTMCell_69329362092464
// MI455X (gfx1250) — compile-verified
//
#include <hip/hip_runtime.h>
#include <math.h>

// ---------------- problem constants ----------------
#define UNITS 512
#define MEMN  16384
#define IN_DIM 512
#define BATCH 64
#define DECAY 0.95f

typedef float v2f __attribute__((ext_vector_type(2)));
typedef float v8f __attribute__((ext_vector_type(8)));

// V_WMMA_F32_16X16X4_F32: D(16x16,f32) = A(16x4,f32) x B(4x16,f32) + C
// A frag per lane: a.x = A[lane&15, 2*(lane>>4)], a.y = A[lane&15, 2*(lane>>4)+1]
// B frag per lane: b.x = B[2*(lane>>4), lane&15], b.y = B[2*(lane>>4)+1, lane&15]
// C/D per lane: c[v] = D[v + 8*(lane>>4), lane&15]
__device__ __forceinline__ v8f wmma4(v2f a, v2f b, v8f c) {
    return __builtin_amdgcn_wmma_f32_16x16x4_f32(
        /*neg_a=*/false, a, /*neg_b=*/false, b,
        /*c_mod=*/(short)0, c, /*reuse_a=*/false, /*reuse_b=*/false);
}

__device__ __forceinline__ float hsig(float x) {
    return fminf(fmaxf(0.2f * x + 0.5f, 0.0f), 1.0f);
}

static __device__ __forceinline__ v8f vzero8() {
    v8f z = {0.f,0.f,0.f,0.f,0.f,0.f,0.f,0.f};
    return z;
}

// ---------------- K1: Z = inputs@kernel + h@RK[:, :4U] + bias ; Ri = r@RK[:,4U:5U]
// jobs: 512 waves for Z tiles (4 mt x 128 nt), 32 waves for Ri tiles (4 mt x 8 nt)
__global__ void k_gemm_z(const float* __restrict__ inp, const float* __restrict__ h_tm1,
                         const float* __restrict__ r_tm1, const float* __restrict__ kern,
                         const float* __restrict__ rk, const float* __restrict__ bias,
                         float* __restrict__ Z, float* __restrict__ Ri) {
    int wave = (blockIdx.x * 256 + threadIdx.x) >> 5;
    int lane = threadIdx.x & 31;
    int lo = lane & 15, hi = lane >> 4;
    if (wave < 512) {
        int mt = wave >> 7;      // batch tile 0..3
        int nt = wave & 127;     // col tile 0..127 (4U = 2048)
        int mrow = mt * 16 + lo;
        int n = nt * 16 + lo;
        v8f acc = vzero8();
        #pragma unroll 4
        for (int kk = 0; kk < 512; kk += 4) {
            int k0 = kk + 2 * hi;
            v2f a = *(const v2f*)(inp + mrow * 512 + k0);
            v2f b; b.x = kern[k0 * 2048 + n]; b.y = kern[(k0 + 1) * 2048 + n];
            acc = wmma4(a, b, acc);
        }
        #pragma unroll 4
        for (int kk = 0; kk < 512; kk += 4) {
            int k0 = kk + 2 * hi;
            v2f a = *(const v2f*)(h_tm1 + mrow * 512 + k0);
            v2f b; b.x = rk[k0 * 2560 + n]; b.y = rk[(k0 + 1) * 2560 + n];
            acc = wmma4(a, b, acc);
        }
        float bv = bias[n];
        #pragma unroll
        for (int v = 0; v < 8; v++) {
            int row = mt * 16 + v + 8 * hi;
            Z[row * 2048 + n] = acc[v] + bv;
        }
    } else {
        int jr = wave - 512;     // 0..31
        int mt = jr >> 3, nt = jr & 7;
        int mrow = mt * 16 + lo;
        int n = nt * 16 + lo;
        v8f acc = vzero8();
        #pragma unroll 4
        for (int kk = 0; kk < 512; kk += 4) {
            int k0 = kk + 2 * hi;
            v2f a = *(const v2f*)(r_tm1 + mrow * 512 + k0);
            v2f b; b.x = rk[k0 * 2560 + 2048 + n]; b.y = rk[(k0 + 1) * 2560 + 2048 + n];
            acc = wmma4(a, b, acc);
        }
        #pragma unroll
        for (int v = 0; v < 8; v++) {
            int row = mt * 16 + v + 8 * hi;
            Ri[row * 512 + n] = acc[v];
        }
    }
}

// ---------------- K1b: elementwise LSTM gates -> h, c
__global__ void k_gates(const float* __restrict__ Z, const float* __restrict__ Ri,
                        const float* __restrict__ c_tm1,
                        float* __restrict__ h_out, float* __restrict__ c_out) {
    int idx = blockIdx.x * 256 + threadIdx.x;   // 32768
    int b = idx >> 9, u = idx & 511;
    const float* zr = Z + b * 2048;
    float zi = zr[u] + Ri[idx];
    float zf = zr[512 + u];
    float zc = zr[1024 + u];
    float zo = zr[1536 + u];
    float ig = hsig(zi), fg = hsig(zf), og = hsig(zo);
    float c = fg * c_tm1[idx] + ig * tanhf(zc);
    h_out[idx] = og * tanhf(c);
    c_out[idx] = c;
}

// ---------------- K2: per-batch key norm: hnorm[b] = rsqrt(max(sum_u h[b,u]^2,1e-12))
__global__ void k_hnorm(const float* __restrict__ h, float* __restrict__ hnorm) {
    __shared__ float s[256];
    int b = blockIdx.x, t = threadIdx.x;
    float v0 = h[b * 512 + t], v1 = h[b * 512 + 256 + t];
    s[t] = v0 * v0 + v1 * v1;
    __syncthreads();
    for (int o = 128; o > 0; o >>= 1) {
        if (t < o) s[t] += s[t + o];
        __syncthreads();
    }
    if (t == 0) hnorm[b] = rsqrtf(fmaxf(s[0], 1e-12f));
}

// ---------------- K3: per-memory-row norm (one wave per row)
__global__ void k_mnorm(const float* __restrict__ m, float* __restrict__ mnorm) {
    int wave = (blockIdx.x * 256 + threadIdx.x) >> 5;   // row 0..16383
    int lane = threadIdx.x & 31;
    const float* row = m + (size_t)wave * 512;
    float ss = 0.f;
    #pragma unroll
    for (int i = 0; i < 4; i++) {
        float4 v = *(const float4*)(row + lane * 4 + i * 128);
        ss += v.x * v.x + v.y * v.y + v.z * v.z + v.w * v.w;
    }
    #pragma unroll
    for (int d = 16; d > 0; d >>= 1) ss += __shfl_xor(ss, d, 32);
    if (lane == 0) mnorm[wave] = rsqrtf(fmaxf(ss, 1e-12f));
}

// ---------------- K4: fused cos GEMM + softmax(axis=batch) + c_wr / c_ww / c_wu
// block = 256 thr = 8 waves; wave handles 16 mem rows x all 64 batch cols; grid = 128
__global__ void k_cos_softmax(const float* __restrict__ m_tm1, const float* __restrict__ h,
                              const float* __restrict__ hnorm, const float* __restrict__ mnorm,
                              const float* __restrict__ c_wr_tm1, const float* __restrict__ c_wlu_tm1,
                              const float* __restrict__ c_wu_tm1, const float* __restrict__ wgate,
                              float* __restrict__ c_wr_out, float* __restrict__ c_wu_out,
                              float* __restrict__ c_ww_out) {
    __shared__ v2f bf[16384];   // 128KB: pre-swizzled B frags [kkIdx(128)][nt(4)][lane(32)]
    int t = threadIdx.x;
    for (int e = t; e < 16384; e += 256) {
        int kkIdx = e >> 7, rem = e & 127;
        int nt = rem >> 5, ln = rem & 31;
        int n = nt * 16 + (ln & 15);                 // batch col
        int k0 = kkIdx * 4 + 2 * (ln >> 4);          // unit (K dim), even
        float hn = hnorm[n];
        v2f hv = *(const v2f*)(h + n * 512 + k0);
        v2f o; o.x = hv.x * hn; o.y = hv.y * hn;
        bf[e] = o;
    }
    __syncthreads();

    int wave = t >> 5, lane = t & 31, lo = lane & 15, hi = lane >> 4;
    int mbase = blockIdx.x * 128 + wave * 16;
    int arow_i = mbase + lo;
    float rn = mnorm[arow_i];
    const float* arow = m_tm1 + (size_t)arow_i * 512 + 2 * hi;

    v8f a0 = vzero8(), a1 = vzero8(), a2 = vzero8(), a3 = vzero8();
    #pragma unroll 4
    for (int kkIdx = 0; kkIdx < 128; kkIdx++) {
        v2f am = *(const v2f*)(arow + kkIdx * 4);
        v2f a; a.x = am.x * rn; a.y = am.y * rn;
        int base = kkIdx * 128 + lane;
        a0 = wmma4(a, bf[base], a0);
        a1 = wmma4(a, bf[base + 32], a1);
        a2 = wmma4(a, bf[base + 64], a2);
        a3 = wmma4(a, bf[base + 96], a3);
    }

    float wg = 1.f / (1.f + expf(-wgate[0]));
    #pragma unroll
    for (int v = 0; v < 8; v++) {
        float x0 = a0[v], x1 = a1[v], x2 = a2[v], x3 = a3[v];
        float mx = fmaxf(fmaxf(x0, x1), fmaxf(x2, x3));
        #pragma unroll
        for (int d = 1; d < 16; d <<= 1) mx = fmaxf(mx, __shfl_xor(mx, d, 32));
        float e0 = expf(x0 - mx), e1 = expf(x1 - mx), e2 = expf(x2 - mx), e3 = expf(x3 - mx);
        float sm = e0 + e1 + e2 + e3;
        #pragma unroll
        for (int d = 1; d < 16; d <<= 1) sm += __shfl_xor(sm, d, 32);
        float inv = 1.f / sm;
        int r = mbase + v + 8 * hi;
        float es[4] = {e0, e1, e2, e3};
        #pragma unroll
        for (int tt = 0; tt < 4; tt++) {
            int idx = r * 64 + tt * 16 + lo;
            float wr = es[tt] * inv;
            float ww = wg * c_wr_tm1[idx] + (1.f - wg) + c_wlu_tm1[idx];
            float wu = DECAY * c_wu_tm1[idx] + wr + ww;
            c_wr_out[idx] = wr;
            c_ww_out[idx] = ww;
            c_wu_out[idx] = wu;
        }
    }
}

// ---------------- K5: read = c_wr^T @ m_tm1, split-K=16 partials (deterministic)
__global__ void k_read_partial(const float* __restrict__ c_wr, const float* __restrict__ m_tm1,
                               float* __restrict__ part) {
    int wave = (blockIdx.x * 256 + threadIdx.x) >> 5;  // 0..2047
    int lane = threadIdx.x & 31, lo = lane & 15, hi = lane >> 4;
    int tile = wave >> 4, s = wave & 15;
    int mt = tile >> 5, nt = tile & 31;
    int bcol = mt * 16 + lo;     // batch (M dim)
    int n = nt * 16 + lo;        // unit (N dim)
    v8f acc = vzero8();
    int kstart = s * 1024;
    #pragma unroll 4
    for (int kk = kstart; kk < kstart + 1024; kk += 4) {
        int k0 = kk + 2 * hi;
        v2f a; a.x = c_wr[k0 * 64 + bcol];           a.y = c_wr[(k0 + 1) * 64 + bcol];
        v2f b; b.x = m_tm1[(size_t)k0 * 512 + n];    b.y = m_tm1[(size_t)(k0 + 1) * 512 + n];
        acc = wmma4(a, b, acc);
    }
    float* dst = part + (size_t)wave * 256 + lane * 8;
    #pragma unroll
    for (int v = 0; v < 8; v++) dst[v] = acc[v];
}

__global__ void k_read_reduce(const float* __restrict__ part, float* __restrict__ read_out) {
    int idx = blockIdx.x * 256 + threadIdx.x;   // 32768
    int b = idx >> 9, u = idx & 511;
    int mt = b >> 4, rrow = b & 15, nt = u >> 4, col = u & 15;
    int lane = col + 16 * (rrow >> 3);
    int v = rrow & 7;
    int tile = mt * 32 + nt;
    float sum = 0.f;
    #pragma unroll
    for (int s = 0; s < 16; s++)
        sum += part[((size_t)(tile * 16 + s)) * 256 + lane * 8 + v];
    read_out[idx] = sum;
}

// ---------------- K6: per-batch-column min & first-argmin of c_wu
__global__ void k_colmin(const float* __restrict__ c_wu, float* __restrict__ minv,
                         int* __restrict__ amin) {
    __shared__ float sv[256];
    __shared__ int si[256];
    int b = blockIdx.x, t = threadIdx.x;
    float best = 3.402823466e38f; int bi = 0;
    for (int m = t; m < MEMN; m += 256) {
        float v = c_wu[m * 64 + b];
        if (v < best) { best = v; bi = m; }         // ascending m: strict < keeps first min
    }
    sv[t] = best; si[t] = bi;
    __syncthreads();
    for (int o = 128; o > 0; o >>= 1) {
        if (t < o) {
            float v2 = sv[t + o]; int i2 = si[t + o];
            if (v2 < sv[t] || (v2 == sv[t] && i2 < si[t])) { sv[t] = v2; si[t] = i2; }
        }
        __syncthreads();
    }
    if (t == 0) { minv[b] = sv[0]; amin[b] = si[0]; }
}

__global__ void k_wlu(const float* __restrict__ c_wu, const float* __restrict__ minv,
                      float* __restrict__ c_wlu) {
    int idx = blockIdx.x * 256 + threadIdx.x;   // 1048576
    int b = idx & 63;
    c_wlu[idx] = (c_wu[idx] <= minv[b]) ? 1.f : 0.f;
}

__global__ void k_keep(const int* __restrict__ amin, float* __restrict__ keep) {
    int m = blockIdx.x * 256 + threadIdx.x;     // 16384
    float kp = 1.f;
    #pragma unroll
    for (int b = 0; b < 64; b++) if (amin[b] == m) kp = 0.f;
    keep[m] = kp;
}

// ---------------- K7: m_out = m_tm1*keep + c_ww @ h   (16384x64 @ 64x512, WMMA)
// block = 8 waves, one mtile per block, wave covers 4 ntiles; grid = 1024
__global__ void k_mem_update(const float* __restrict__ c_ww, const float* __restrict__ h,
                             const float* __restrict__ m_tm1, const float* __restrict__ keep,
                             float* __restrict__ m_out) {
    int t = threadIdx.x;
    int wave = t >> 5, lane = t & 31, lo = lane & 15, hi = lane >> 4;
    int mbase = blockIdx.x * 16;
    const float* arow = c_ww + (mbase + lo) * 64 + 2 * hi;
    int ntb = wave * 4;
    v8f acc0 = vzero8(), acc1 = vzero8(), acc2 = vzero8(), acc3 = vzero8();
    #pragma unroll
    for (int kkIdx = 0; kkIdx < 16; kkIdx++) {
        v2f a = *(const v2f*)(arow + kkIdx * 4);
        int k0 = kkIdx * 4 + 2 * hi;
        int n0 = ntb * 16 + lo;
        v2f b;
        b.x = h[k0 * 512 + n0];      b.y = h[(k0 + 1) * 512 + n0];      acc0 = wmma4(a, b, acc0);
        b.x = h[k0 * 512 + n0 + 16]; b.y = h[(k0 + 1) * 512 + n0 + 16]; acc1 = wmma4(a, b, acc1);
        b.x = h[k0 * 512 + n0 + 32]; b.y = h[(k0 + 1) * 512 + n0 + 32]; acc2 = wmma4(a, b, acc2);
        b.x = h[k0 * 512 + n0 + 48]; b.y = h[(k0 + 1) * 512 + n0 + 48]; acc3 = wmma4(a, b, acc3);
    }
    #pragma unroll
    for (int v = 0; v < 8; v++) {
        int r = mbase + v + 8 * hi;
        float kp = keep[r];
        size_t base = (size_t)r * 512 + ntb * 16 + lo;
        m_out[base]      = m_tm1[base]      * kp + acc0[v];
        m_out[base + 16] = m_tm1[base + 16] * kp + acc1[v];
        m_out[base + 32] = m_tm1[base + 32] * kp + acc2[v];
        m_out[base + 48] = m_tm1[base + 48] * kp + acc3[v];
    }
}

// ---------------- launch ----------------
extern "C" void kernel_launch(void* const* d_in, const int* in_sizes, int n_in,
                              void* d_out, int out_size, void* d_ws, size_t ws_size,
                              hipStream_t stream) {
    const float* inputs    = (const float*)d_in[0];
    const float* h_tm1     = (const float*)d_in[1];
    const float* c_tm1     = (const float*)d_in[2];
    const float* r_tm1     = (const float*)d_in[3];
    const float* m_tm1     = (const float*)d_in[4];
    const float* c_wu_tm1  = (const float*)d_in[5];
    const float* c_wlu_tm1 = (const float*)d_in[6];
    const float* c_wr_tm1  = (const float*)d_in[7];
    const float* kern      = (const float*)d_in[8];
    const float* rk        = (const float*)d_in[9];
    const float* bias      = (const float*)d_in[10];
    const float* wgate     = (const float*)d_in[11];

    float* out = (float*)d_out;
    float* h_out    = out;                                  // 64*512
    float* c_out    = out + 32768;                          // 64*512
    float* read_out = out + 65536;                          // 64*512
    float* m_out    = out + 98304;                          // 16384*512
    float* cwu_out  = out + 98304 + 8388608;                // 16384*64
    float* cwlu_out = cwu_out + 1048576;
    float* cwr_out  = cwlu_out + 1048576;

    float* ws = (float*)d_ws;                               // needs ~6.8 MB
    float* Z     = ws;                  // 131072
    float* Ri    = ws + 131072;         // 32768
    float* hnorm = ws + 163840;         // 64
    float* mnorm = ws + 163904;         // 16384
    float* c_ww  = ws + 180288;         // 1048576
    float* minv  = ws + 1228864;        // 64
    int*   amin  = (int*)(ws + 1228928);// 64
    float* keep  = ws + 1228992;        // 16384
    float* part  = ws + 1245376;        // 524288

    // LSTM gates
    k_gemm_z<<<68, 256, 0, stream>>>(inputs, h_tm1, r_tm1, kern, rk, bias, Z, Ri);
    k_gates<<<128, 256, 0, stream>>>(Z, Ri, c_tm1, h_out, c_out);
    // norms
    k_hnorm<<<64, 256, 0, stream>>>(h_out, hnorm);
    k_mnorm<<<2048, 256, 0, stream>>>(m_tm1, mnorm);
    // fused cosine GEMM + softmax + c_wr/c_ww/c_wu
    k_cos_softmax<<<128, 256, 0, stream>>>(m_tm1, h_out, hnorm, mnorm,
                                           c_wr_tm1, c_wlu_tm1, c_wu_tm1, wgate,
                                           cwr_out, cwu_out, c_ww);
    // read = c_wr^T @ m_tm1 (split-K=16, deterministic reduce)
    k_read_partial<<<256, 256, 0, stream>>>(cwr_out, m_tm1, part);
    k_read_reduce<<<128, 256, 0, stream>>>(part, read_out);
    // least-used bookkeeping
    k_colmin<<<64, 256, 0, stream>>>(cwu_out, minv, amin);
    k_wlu<<<4096, 256, 0, stream>>>(cwu_out, minv, cwlu_out);
    k_keep<<<64, 256, 0, stream>>>(amin, keep);
    // memory rewrite
    k_mem_update<<<1024, 256, 0, stream>>>(c_ww, h_out, m_tm1, keep, m_out);
}